// CrossViewGCN_Layer1_46634754900288
// MI455X (gfx1250) — compile-verified
//
#include <hip/hip_runtime.h>

typedef float v2f __attribute__((ext_vector_type(2)));
typedef float v8f __attribute__((ext_vector_type(8)));

constexpr int kNSmp = 8192;           // NSMP
constexpr int kNA   = 512;            // NA
constexpr int kDOut = 256;            // DOUT
constexpr int kInW  = 520 + 8192;     // 8712 input row pitch
constexpr int kOutW = kDOut + kNSmp;  // 8448 output row pitch
constexpr int kRowsPerBlk = 128;
constexpr int kRowTiles   = kNSmp / kRowsPerBlk;  // 64

// ---------------------------------------------------------------------------
// Kernel 1: stream A once. Write (A + I) into out[:, 256:] and emit partial
// column sums (per 128-row tile) for the degree vector. No atomics ->
// bitwise deterministic across replays.
// ---------------------------------------------------------------------------
__global__ __launch_bounds__(256) void colsum_copy(const float* __restrict__ in,
                                                   float* __restrict__ out,
                                                   float* __restrict__ partial) {
  const int c  = blockIdx.x * 256 + threadIdx.x;   // column 0..8191
  const int r0 = blockIdx.y * kRowsPerBlk;
  const float* __restrict__ A = in + kNA;          // A starts at col 512
  float acc = 0.0f;
#pragma unroll 4
  for (int r = 0; r < kRowsPerBlk; ++r) {
    const int row = r0 + r;
    float v = A[(size_t)row * kInW + c];
    v += (row == c) ? 1.0f : 0.0f;                 // A + I
    acc += v;
    out[(size_t)row * kOutW + kDOut + c] = v;      // right block of output
  }
  partial[(size_t)blockIdx.y * kNSmp + c] = acc;
}

// ---------------------------------------------------------------------------
// Kernel 2: d[c] = sum of 64 partials; dis[c] = rsqrt(d[c])
// ---------------------------------------------------------------------------
__global__ __launch_bounds__(256) void reduce_rsqrt(const float* __restrict__ partial,
                                                    float* __restrict__ dis) {
  const int c = blockIdx.x * 256 + threadIdx.x;
  float s = 0.0f;
#pragma unroll 8
  for (int t = 0; t < kRowTiles; ++t) s += partial[(size_t)t * kNSmp + c];
  dis[c] = rsqrtf(s);
}

// ---------------------------------------------------------------------------
// f32 WMMA GEMM:  C[m,n] = dis[m] * ( sum_k A[m,k]*B[k,n]  (+ Sid[m,n]) )
// Block tile: M=32, N=256, K-chunk=16. 256 threads = 8 waves arranged 2x4;
// each wave owns a 16x64 slab = 4 accumulators of V_WMMA_F32_16X16X4_F32.
//
// Tiles are staged with gfx1250 GLOBAL_LOAD_ASYNC_TO_LDS (ASYNCcnt) into a
// double-buffered LDS pipeline; the A tile is transposed to [k][m] by giving
// each lane its own LDS destination address (per-lane async scatter).
// ---------------------------------------------------------------------------
template <bool kAddIdent>
__global__ __launch_bounds__(256) void gemm_wmma_f32(const float* __restrict__ A, int lda,
                                                     const float* __restrict__ B, int ldb,
                                                     float* __restrict__ C, int ldc,
                                                     const float* __restrict__ dis,
                                                     const float* __restrict__ Sid,
                                                     int K) {
  __shared__ float sA[2][16][32];    // [buf][k][m]   2 x 2 KB
  __shared__ float sB[2][16][256];   // [buf][k][n]   2 x 16 KB

  const int tid  = threadIdx.x;
  const int lane = tid & 31;
  const int wave = tid >> 5;
  const int wr   = wave >> 2;     // 0..1 : wave row-slab (16 rows each)
  const int wc   = wave & 3;      // 0..3 : wave col-slab (64 cols each)
  const int h    = lane >> 4;     // half-wave
  const int l15  = lane & 15;

  const int m0 = blockIdx.x * 32;
  const int n0 = blockIdx.y * 256;

  // A-tile loader: 32 rows x 16 k, two b32 asyncs per thread (transpose scatter)
  const int ar = tid >> 3;          // 0..31  (m within tile)
  const int ac = (tid & 7) << 1;    // 0,2,..,14 (k within chunk)
  // B-tile loader: 16 rows x 256 n, four b128 asyncs per thread
  const int br = tid >> 6;          // 0..3
  const int bc = (tid & 63) << 2;   // 0..252

  const float* __restrict__ gA = &A[(size_t)(m0 + ar) * lda + ac];  // + k0
  const float* __restrict__ gB = &B[(size_t)br * ldb + n0 + bc];    // + k0*ldb

  // Issue one chunk's async tile loads into LDS buffer `buf`.
  auto issue_chunk = [&](int k0, int buf) {
    const unsigned ldsA0 = (unsigned)(uintptr_t)&sA[buf][ac + 0][ar];
    const unsigned ldsA1 = (unsigned)(uintptr_t)&sA[buf][ac + 1][ar];
    const float* ga = gA + k0;
    asm volatile("global_load_async_to_lds_b32 %0, %1, off"
                 :: "v"(ldsA0), "v"(ga) : "memory");
    asm volatile("global_load_async_to_lds_b32 %0, %1, off"
                 :: "v"(ldsA1), "v"(ga + 1) : "memory");
#pragma unroll
    for (int p = 0; p < 4; ++p) {
      const unsigned ldsB = (unsigned)(uintptr_t)&sB[buf][br + 4 * p][bc];
      const float* gb = gB + (size_t)(k0 + 4 * p) * ldb;
      asm volatile("global_load_async_to_lds_b128 %0, %1, off"
                   :: "v"(ldsB), "v"(gb) : "memory");
    }
  };

  v8f acc[4] = {};

  issue_chunk(0, 0);
  int buf = 0;
  for (int k0 = 0; k0 < K; k0 += 16) {
    // Drain this wave's outstanding async-LDS loads, then make the tile
    // visible to (and protect the other buffer from) all waves.
    asm volatile("s_wait_asynccnt 0x0" ::: "memory");
    __syncthreads();

    if (k0 + 16 < K) issue_chunk(k0 + 16, buf ^ 1);  // overlap with compute

#pragma unroll
    for (int ko = 0; ko < 16; ko += 4) {
      // A fragment 16x4: lane l -> (M = l&15, K = 2*(l>>4) + {0,1})
      v2f a;
      a.x = sA[buf][ko + 2 * h + 0][wr * 16 + l15];
      a.y = sA[buf][ko + 2 * h + 1][wr * 16 + l15];
#pragma unroll
      for (int t = 0; t < 4; ++t) {
        // B fragment 4x16: lane l -> (K = 2*(l>>4) + {0,1}, N = l&15)
        v2f b;
        b.x = sB[buf][ko + 2 * h + 0][wc * 64 + t * 16 + l15];
        b.y = sB[buf][ko + 2 * h + 1][wc * 64 + t * 16 + l15];
        acc[t] = __builtin_amdgcn_wmma_f32_16x16x4_f32(
            /*neg_a=*/false, a, /*neg_b=*/false, b,
            /*c_mod=*/(short)0, acc[t], /*reuse_a=*/false, /*reuse_b=*/false);
      }
    }
    buf ^= 1;
  }

  // Epilogue: C/D layout -> VGPR i holds row i + 8*h, col = l15
#pragma unroll
  for (int t = 0; t < 4; ++t) {
    const int gcol = n0 + wc * 64 + t * 16 + l15;
#pragma unroll
    for (int i = 0; i < 8; ++i) {
      const int grow = m0 + wr * 16 + 8 * h + i;
      float v = acc[t][i];
      if constexpr (kAddIdent) v += Sid[(size_t)grow * kDOut + gcol];  // +I term
      C[(size_t)grow * ldc + gcol] = dis[grow] * v;
    }
  }
}

// ---------------------------------------------------------------------------
// Launch: out = concat( dis∘(A+I)@(dis∘(X@W)) , A+I )
// ---------------------------------------------------------------------------
extern "C" void kernel_launch(void* const* d_in, const int* in_sizes, int n_in,
                              void* d_out, int out_size, void* d_ws, size_t ws_size,
                              hipStream_t stream) {
  const float* in = (const float*)d_in[0];   // (8192, 8712)
  const float* w  = (const float*)d_in[1];   // (512, 256)
  float* out = (float*)d_out;                // (8192, 8448)
  float* ws  = (float*)d_ws;

  float* partial = ws;                                   // 64 * 8192
  float* dis     = ws + (size_t)kRowTiles * kNSmp;       // 8192
  float* Sp      = dis + kNSmp;                          // 8192 * 256

  // 1) stream A: write A+I to out[:,256:], partial column sums
  colsum_copy<<<dim3(kNSmp / 256, kRowTiles), 256, 0, stream>>>(in, out, partial);

  // 2) dis = rsqrt(colsum(A+I))
  reduce_rsqrt<<<kNSmp / 256, 256, 0, stream>>>(partial, dis);

  // 3) S' = dis ∘rows (X @ W)        M=8192 K=512 N=256
  gemm_wmma_f32<false><<<dim3(kNSmp / 32, kDOut / 256), 256, 0, stream>>>(
      in, kInW, w, kDOut, Sp, kDOut, dis, nullptr, kNA);

  // 4) out[:, :256] = dis ∘rows ( A @ S' + S' )   M=8192 K=8192 N=256
  gemm_wmma_f32<true><<<dim3(kNSmp / 32, kDOut / 256), 256, 0, stream>>>(
      in + kNA, kInW, Sp, kDOut, out, kOutW, dis, Sp, kNSmp);
}